// GAT_41025527611679
// MI455X (gfx1250) — compile-verified
//
#include <hip/hip_runtime.h>
#include <math.h>

// ---- problem constants (match reference) ----
#define NN   50000          // nodes (3125 * 16 exactly)
#define NE   800000         // edges (plus NN self loops)
#define FIN  128
#define C1   100
#define C1P  112            // C1 padded to 7 * 16 for WMMA tiling
#define C2   4
#define C2P  16
#define NEG_SLOPE 0.2f

typedef float v2f __attribute__((ext_vector_type(2)));
typedef float v8f __attribute__((ext_vector_type(8)));

// ---------------------------------------------------------------------------
// init: zero layer-1 accumulator (N x C1P), layer-2 accumulator (N x 4),
// denominators, and set segment-max buffers to -inf.
// ---------------------------------------------------------------------------
__global__ void gat_init(float* __restrict__ h1b, float* __restrict__ out2,
                         float* __restrict__ d1, float* __restrict__ d2,
                         float* __restrict__ m1, float* __restrict__ m2) {
  const int stride = gridDim.x * blockDim.x;
  const int total = NN * C1P;
  for (int i = blockIdx.x * blockDim.x + threadIdx.x; i < total; i += stride) {
    h1b[i] = 0.f;
    if (i < NN * C2) out2[i] = 0.f;
    if (i < NN) { d1[i] = 0.f; d2[i] = 0.f; m1[i] = -INFINITY; m2[i] = -INFINITY; }
  }
}

// ---------------------------------------------------------------------------
// weight pre-pad (transpose + zero-pad) so GEMM inner loops are guard-free and
// each B fragment is a single aligned global_load_b64.
// Wp1[col][k] = W1[k][col] for col<100 else 0         (112 x 128)
// Wp2[col][k] = W2[k][col] for col<4 && k<100 else 0  (16 x 112)
// ---------------------------------------------------------------------------
__global__ void gat_pad_w1(const float* __restrict__ W, float* __restrict__ Wp) {
  const int i = blockIdx.x * blockDim.x + threadIdx.x;
  if (i >= C1P * FIN) return;
  const int col = i / FIN, k = i % FIN;
  Wp[i] = (col < C1) ? W[(size_t)k * C1 + col] : 0.f;
}

__global__ void gat_pad_w2(const float* __restrict__ W, float* __restrict__ Wp) {
  const int i = blockIdx.x * blockDim.x + threadIdx.x;
  if (i >= C2P * C1P) return;
  const int col = i / C1P, k = i % C1P;
  Wp[i] = (col < C2 && k < C1) ? W[(size_t)k * C2 + col] : 0.f;
}

// ---------------------------------------------------------------------------
// GEMM1: h1[N, C1P] = x[N, FIN] @ W1[FIN, C1] via V_WMMA_F32_16X16X4_F32.
// One wave per 16x16 tile. All 32 A and 32 B fragments preloaded to VGPRs,
// then 32 back-to-back WMMAs on the accumulate chain.
// A frag (16x4 f32): lane<16 -> M=lane, K={k,k+1}; lane>=16 -> K={k+2,k+3}
// C/D frag: VGPR r -> M = 16*tm + r + 8*(lane>=16), N = 16*tn + (lane&15)
// ---------------------------------------------------------------------------
__global__ void gat_gemm1(const float* __restrict__ x, const float* __restrict__ Wp,
                          float* __restrict__ h1) {
  const int wid  = (blockIdx.x * blockDim.x + threadIdx.x) >> 5;
  const int lane = threadIdx.x & 31;
  const int TN = C1P / 16;                     // 7 tiles along N
  const int tm = wid / TN;
  const int tn = wid % TN;
  if (tm >= NN / 16) return;                   // wave-uniform

  const bool hi  = lane >= 16;
  const int  row = tm * 16 + (lane & 15);
  const int  col = tn * 16 + (lane & 15);
  const float* arow = x  + (size_t)row * FIN + (hi ? 2 : 0);
  const float* bcol = Wp + (size_t)col * FIN + (hi ? 2 : 0);

  v2f a[32], b[32];
  #pragma unroll
  for (int i = 0; i < 32; ++i) {
    a[i] = *(const v2f*)(arow + 4 * i);
    b[i] = *(const v2f*)(bcol + 4 * i);
  }
  v8f c = {};
  #pragma unroll
  for (int i = 0; i < 32; ++i)
    c = __builtin_amdgcn_wmma_f32_16x16x4_f32(false, a[i], false, b[i],
                                              (short)0, c, false, false);

  const int mbase = tm * 16 + (hi ? 8 : 0);
  #pragma unroll
  for (int r = 0; r < 8; ++r)
    h1[(size_t)(mbase + r) * C1P + col] = c[r];
}

// ---------------------------------------------------------------------------
// GEMM2: h2[N, 4] = h1b[N, C1P] @ W2[C1, C2]  (K padded to 112; pad cols of
// h1b are zero by construction, Wp2 pre-padded). One wave per 16-row tile.
// ---------------------------------------------------------------------------
__global__ void gat_gemm2(const float* __restrict__ h1b, const float* __restrict__ Wp,
                          float* __restrict__ h2) {
  const int wid  = (blockIdx.x * blockDim.x + threadIdx.x) >> 5;
  const int lane = threadIdx.x & 31;
  if (wid >= NN / 16) return;                  // wave-uniform

  const bool hi  = lane >= 16;
  const int  row = wid * 16 + (lane & 15);
  const int  col = lane & 15;
  const float* arow = h1b + (size_t)row * C1P + (hi ? 2 : 0);
  const float* bcol = Wp  + (size_t)col * C1P + (hi ? 2 : 0);

  v2f a[28], b[28];
  #pragma unroll
  for (int i = 0; i < 28; ++i) {
    a[i] = *(const v2f*)(arow + 4 * i);
    b[i] = *(const v2f*)(bcol + 4 * i);
  }
  v8f c = {};
  #pragma unroll
  for (int i = 0; i < 28; ++i)
    c = __builtin_amdgcn_wmma_f32_16x16x4_f32(false, a[i], false, b[i],
                                              (short)0, c, false, false);

  const int mbase = wid * 16 + (hi ? 8 : 0);
  if (col < C2) {
    #pragma unroll
    for (int r = 0; r < 8; ++r)
      h2[(size_t)(mbase + r) * C2 + col] = c[r];
  }
}

// ---------------------------------------------------------------------------
// attention dots, layer 1: one wave per node, coalesced row read + xor reduce
// ---------------------------------------------------------------------------
__global__ void gat_dots1(const float* __restrict__ h1, const float* __restrict__ as,
                          const float* __restrict__ ad,
                          float* __restrict__ a_src, float* __restrict__ a_dst) {
  const int wid  = (blockIdx.x * blockDim.x + threadIdx.x) >> 5;
  const int lane = threadIdx.x & 31;
  if (wid >= NN) return;
  const float* row = h1 + (size_t)wid * C1P;
  float s = 0.f, d = 0.f;
  for (int ch = lane; ch < C1; ch += 32) {
    const float v = row[ch];
    s += v * as[ch];
    d += v * ad[ch];
  }
  #pragma unroll
  for (int off = 16; off > 0; off >>= 1) {
    s += __shfl_xor(s, off, 32);
    d += __shfl_xor(d, off, 32);
  }
  if (lane == 0) { a_src[wid] = s; a_dst[wid] = d; }
}

// attention dots, layer 2 (C=4): one thread per node
__global__ void gat_dots2(const float* __restrict__ h2, const float* __restrict__ as,
                          const float* __restrict__ ad,
                          float* __restrict__ a_src, float* __restrict__ a_dst) {
  const int n = blockIdx.x * blockDim.x + threadIdx.x;
  if (n >= NN) return;
  const float4 h = *(const float4*)(h2 + (size_t)n * C2);
  a_src[n] = h.x * as[0] + h.y * as[1] + h.z * as[2] + h.w * as[3];
  a_dst[n] = h.x * ad[0] + h.y * ad[1] + h.z * ad[2] + h.w * ad[3];
}

// ---------------------------------------------------------------------------
// edge softmax passes (shared by both layers; self-loops appended at e >= NE)
// ---------------------------------------------------------------------------
__device__ __forceinline__ void atomicMaxF32(float* addr, float val) {
  if (val >= 0.f) atomicMax((int*)addr, __float_as_int(val));
  else            atomicMin((unsigned int*)addr, __float_as_uint(val));
}

__device__ __forceinline__ void edge_sd(const int* ei, int e, int& s, int& d) {
  if (e < NE) { s = ei[e]; d = ei[NE + e]; }
  else        { s = d = e - NE; }
}

__global__ void gat_edge_max(const int* __restrict__ ei, const float* __restrict__ a_src,
                             const float* __restrict__ a_dst,
                             float* __restrict__ ev, float* __restrict__ m) {
  const int e = blockIdx.x * blockDim.x + threadIdx.x;
  if (e >= NE + NN) return;
  int s, d; edge_sd(ei, e, s, d);
  float v = a_src[s] + a_dst[d];
  v = (v > 0.f) ? v : NEG_SLOPE * v;
  ev[e] = v;
  atomicMaxF32(&m[d], v);
}

__global__ void gat_edge_expsum(const int* __restrict__ ei, float* __restrict__ ev,
                                const float* __restrict__ m, float* __restrict__ denom) {
  const int e = blockIdx.x * blockDim.x + threadIdx.x;
  if (e >= NE + NN) return;
  int s, d; edge_sd(ei, e, s, d);
  const float w = expf(ev[e] - m[d]);
  ev[e] = w;
  atomicAdd(&denom[d], w);
}

// layer-1 aggregation: one wave per edge, lanes cover the 100 channels
__global__ void gat_edge_agg1(const int* __restrict__ ei, const float* __restrict__ ev,
                              const float* __restrict__ denom,
                              const float* __restrict__ h1, float* __restrict__ out) {
  const int wid  = (blockIdx.x * blockDim.x + threadIdx.x) >> 5;
  const int lane = threadIdx.x & 31;
  if (wid >= NE + NN) return;
  int s, d; edge_sd(ei, wid, s, d);
  const float alpha = ev[wid] / denom[d];
  const float* hs = h1 + (size_t)s * C1P;
  float* od = out + (size_t)d * C1P;
  for (int c = lane; c < C1; c += 32)
    atomicAdd(&od[c], hs[c] * alpha);
}

// layer-2 aggregation: one thread per edge, 4 channels
__global__ void gat_edge_agg2(const int* __restrict__ ei, const float* __restrict__ ev,
                              const float* __restrict__ denom,
                              const float* __restrict__ h2, float* __restrict__ out) {
  const int e = blockIdx.x * blockDim.x + threadIdx.x;
  if (e >= NE + NN) return;
  int s, d; edge_sd(ei, e, s, d);
  const float alpha = ev[e] / denom[d];
  const float4 h = *(const float4*)(h2 + (size_t)s * C2);
  float* od = out + (size_t)d * C2;
  atomicAdd(&od[0], h.x * alpha);
  atomicAdd(&od[1], h.y * alpha);
  atomicAdd(&od[2], h.z * alpha);
  atomicAdd(&od[3], h.w * alpha);
}

// layer-1 finalize in place: h1b = relu(accum + b1); pad cols stay zero
__global__ void gat_fin1(float* __restrict__ h1b, const float* __restrict__ b1) {
  const int i = blockIdx.x * blockDim.x + threadIdx.x;
  if (i >= NN * C1P) return;
  const int c = i % C1P;
  if (c < C1) {
    const float v = h1b[i] + b1[c];
    h1b[i] = v > 0.f ? v : 0.f;
  }
}

// layer-2 finalize: y = log_softmax(accum + b2) over 4 classes
__global__ void gat_lsm(const float* __restrict__ out2, const float* __restrict__ b2,
                        float* __restrict__ y) {
  const int n = blockIdx.x * blockDim.x + threadIdx.x;
  if (n >= NN) return;
  float z[C2], mx = -INFINITY;
  #pragma unroll
  for (int c = 0; c < C2; ++c) {
    z[c] = out2[(size_t)n * C2 + c] + b2[c];
    mx = fmaxf(mx, z[c]);
  }
  float sum = 0.f;
  #pragma unroll
  for (int c = 0; c < C2; ++c) sum += expf(z[c] - mx);
  const float lse = logf(sum) + mx;
  #pragma unroll
  for (int c = 0; c < C2; ++c) y[(size_t)n * C2 + c] = z[c] - lse;
}

// ---------------------------------------------------------------------------
extern "C" void kernel_launch(void* const* d_in, const int* in_sizes, int n_in,
                              void* d_out, int out_size, void* d_ws, size_t ws_size,
                              hipStream_t stream) {
  const float* x    = (const float*)d_in[0];
  const int*   ei   = (const int*)  d_in[1];
  const float* W1   = (const float*)d_in[2];
  const float* as1  = (const float*)d_in[3];
  const float* ad1  = (const float*)d_in[4];
  const float* b1   = (const float*)d_in[5];
  const float* W2   = (const float*)d_in[6];
  const float* as2  = (const float*)d_in[7];
  const float* ad2  = (const float*)d_in[8];
  const float* b2   = (const float*)d_in[9];
  float* out = (float*)d_out;

  // workspace carve-up (floats)
  float* w = (float*)d_ws;
  float* h1    = w; w += (size_t)NN * C1P;
  float* h1b   = w; w += (size_t)NN * C1P;   // accum -> relu(accum+b1) in place
  float* asrc1 = w; w += NN;
  float* adst1 = w; w += NN;
  float* m1    = w; w += NN;
  float* d1    = w; w += NN;
  float* h2    = w; w += (size_t)NN * C2;
  float* out2  = w; w += (size_t)NN * C2;
  float* asrc2 = w; w += NN;
  float* adst2 = w; w += NN;
  float* m2    = w; w += NN;
  float* d2v   = w; w += NN;
  float* ev    = w; w += (size_t)(NE + NN);  // reused by both layers
  float* Wp1   = w; w += (size_t)C1P * FIN;  // transposed, zero-padded W1
  float* Wp2   = w; w += (size_t)C2P * C1P;  // transposed, zero-padded W2

  const int NEDGE = NE + NN;
  const int B = 256;

  gat_init<<<4096, B, 0, stream>>>(h1b, out2, d1, d2v, m1, m2);
  gat_pad_w1<<<(C1P * FIN + B - 1) / B, B, 0, stream>>>(W1, Wp1);
  gat_pad_w2<<<(C2P * C1P + B - 1) / B, B, 0, stream>>>(W2, Wp2);

  // layer 1
  {
    const int waves = (NN / 16) * (C1P / 16);
    gat_gemm1<<<(waves * 32 + B - 1) / B, B, 0, stream>>>(x, Wp1, h1);
  }
  gat_dots1<<<(NN * 32 + B - 1) / B, B, 0, stream>>>(h1, as1, ad1, asrc1, adst1);
  gat_edge_max<<<(NEDGE + B - 1) / B, B, 0, stream>>>(ei, asrc1, adst1, ev, m1);
  gat_edge_expsum<<<(NEDGE + B - 1) / B, B, 0, stream>>>(ei, ev, m1, d1);
  gat_edge_agg1<<<((size_t)NEDGE * 32 + B - 1) / B, B, 0, stream>>>(ei, ev, d1, h1, h1b);
  gat_fin1<<<(NN * C1P + B - 1) / B, B, 0, stream>>>(h1b, b1);

  // layer 2
  {
    const int waves = NN / 16;
    gat_gemm2<<<(waves * 32 + B - 1) / B, B, 0, stream>>>(h1b, Wp2, h2);
  }
  gat_dots2<<<(NN + B - 1) / B, B, 0, stream>>>(h2, as2, ad2, asrc2, adst2);
  gat_edge_max<<<(NEDGE + B - 1) / B, B, 0, stream>>>(ei, asrc2, adst2, ev, m2);
  gat_edge_expsum<<<(NEDGE + B - 1) / B, B, 0, stream>>>(ei, ev, m2, d2v);
  gat_edge_agg2<<<(NEDGE + B - 1) / B, B, 0, stream>>>(ei, ev, d2v, h2, out2);
  gat_lsm<<<(NN + B - 1) / B, B, 0, stream>>>(out2, b2, out);
}